// Decoder_9740985827869
// MI455X (gfx1250) — compile-verified
//
#include <hip/hip_runtime.h>
#include <math.h>

// Problem constants (match reference)
#define BB 128     // batch
#define PP 49      // spatial positions
#define DD 512     // hidden / embed / att dim
#define VV 10000   // vocab
#define TT 50      // caption length
#define NS 49      // decode steps = T-1

typedef __attribute__((ext_vector_type(2))) float v2f;
typedef __attribute__((ext_vector_type(8))) float v8f;

enum { ACT_NONE = 0, ACT_RELU = 1, ACT_TANH = 2 };

__device__ __forceinline__ v2f ld2(const float* p) {
  return *reinterpret_cast<const v2f*>(p);
}
__device__ __forceinline__ float sigm(float x) { return 1.f / (1.f + expf(-x)); }

// ---------------------------------------------------------------------------
// Generic 3-segment fp32 WMMA GEMM:
//   C[M,N] = act( sum_s A_s[M,K_s] @ W_s[N,K_s]^T + bias0 + bias1 )
// Wave tile: 16(M) x 64(N); block = 8 waves = 32(M) x 256(N).
// Uses V_WMMA_F32_16X16X4_F32. Requires M % 32 == 0, K_s % 4 == 0, N % 16 == 0.
// f32 WMMA fragment layout (CDNA5 ISA 7.12.2):
//   A 16x4 : lane r=lane%16 -> M=m0+r ; half=lane/16, VGPR v -> K = 2*half+v
//   B 4x16 : lane r -> N=n0+r ; half, VGPR v -> K = 2*half+v  (B[k][n]=W[n][k])
//   C 16x16: VGPR d, half -> M = m0 + d + 8*half ; lane r -> N
// ---------------------------------------------------------------------------
__global__ __launch_bounds__(256) void wmma_gemm3(
    const float* __restrict__ A0, int lda0, const float* __restrict__ W0, int ldw0, int K0,
    const float* __restrict__ A1, int lda1, const float* __restrict__ W1, int ldw1, int K1,
    const float* __restrict__ A2, int lda2, const float* __restrict__ W2, int ldw2, int K2,
    const float* __restrict__ bias0, const float* __restrict__ bias1,
    float* __restrict__ C, long long ldc, int N, int act)
{
  const int lane = threadIdx.x & 31;
  const int wave = threadIdx.x >> 5;
  const int wm = wave >> 2;           // 0..1
  const int wn = wave & 3;            // 0..3
  const int hl = lane >> 4;           // half-wave 0/1
  const int r  = lane & 15;
  const int m0 = blockIdx.y * 32 + wm * 16;
  const int n0 = blockIdx.x * 256 + wn * 64;

  v8f acc[4];
#pragma unroll
  for (int j = 0; j < 4; ++j)
#pragma unroll
    for (int e = 0; e < 8; ++e) acc[j][e] = 0.f;

  const float* As[3]  = {A0, A1, A2};
  const float* Wsp[3] = {W0, W1, W2};
  const int ldas[3] = {lda0, lda1, lda2};
  const int ldws[3] = {ldw0, ldw1, ldw2};
  const int Ks[3]   = {K0, K1, K2};

  for (int s = 0; s < 3; ++s) {
    const int K = Ks[s];
    if (K == 0) continue;
    const float* __restrict__ aRow = As[s] + (long long)(m0 + r) * ldas[s] + 2 * hl;
    const float* wRow[4];
#pragma unroll
    for (int j = 0; j < 4; ++j) {
      int n = n0 + 16 * j + r;
      if (n > N - 1) n = N - 1;       // clamp: keeps loads in-bounds, EXEC all-ones
      wRow[j] = Wsp[s] + (long long)n * ldws[s] + 2 * hl;
      __builtin_prefetch(wRow[j], 0, 0);
    }
    for (int k = 0; k < K; k += 4) {
      v2f a = ld2(aRow + k);
#pragma unroll
      for (int j = 0; j < 4; ++j) {
        v2f bf = ld2(wRow[j] + k);
        acc[j] = __builtin_amdgcn_wmma_f32_16x16x4_f32(
            /*neg_a=*/false, a, /*neg_b=*/false, bf,
            /*c_mod=*/(short)0, acc[j], /*reuse_a=*/false, /*reuse_b=*/false);
      }
    }
  }

#pragma unroll
  for (int j = 0; j < 4; ++j) {
    if (n0 + 16 * j >= N) continue;   // wave-uniform skip (N % 16 == 0)
    const int nc = n0 + 16 * j + r;
    float bv = 0.f;
    if (bias0) bv += bias0[nc];
    if (bias1) bv += bias1[nc];
#pragma unroll
    for (int d = 0; d < 8; ++d) {
      float v = acc[j][d] + bv;
      if (act == ACT_RELU)      v = fmaxf(v, 0.f);
      else if (act == ACT_TANH) v = tanhf(v);
      C[(long long)(m0 + d + 8 * hl) * ldc + nc] = v;
    }
  }
}

// ---------------------------------------------------------------------------
// LSTM cell elementwise (+ optional sentinel): gates layout [i|f|g|o] of D each
// ---------------------------------------------------------------------------
__global__ __launch_bounds__(256) void lstm_kernel(
    const float* __restrict__ gates,   // (B, 4D)
    float* __restrict__ cst,           // (B, D) in-place c state
    float* __restrict__ hst,           // (B, D) h state (written)
    const float* __restrict__ slin,    // (B, D) sentinel pre-activation or null
    float* __restrict__ sout)          // (B, D) sentinel out
{
  int idx = blockIdx.x * 256 + threadIdx.x;
  if (idx >= BB * DD) return;
  int b = idx >> 9;                    // D = 512
  int d = idx & 511;
  const float* g = gates + (long long)b * 4 * DD;
  float ii = sigm(g[d]);
  float ff = sigm(g[DD + d]);
  float gg = tanhf(g[2 * DD + d]);
  float oo = sigm(g[3 * DD + d]);
  float c  = ff * cst[idx] + ii * gg;
  float tc = tanhf(c);
  cst[idx] = c;
  hst[idx] = oo * tc;
  if (slin) sout[idx] = sigm(slin[idx]) * tc;
}

// ---------------------------------------------------------------------------
// Adaptive attention: z -> softmax(50) -> context; one block per batch row
// ---------------------------------------------------------------------------
__global__ __launch_bounds__(256) void attn_kernel(
    const float* __restrict__ vatt,    // (B, P, A)
    const float* __restrict__ hid,     // (B, A)
    const float* __restrict__ sen,     // (B, A)
    const float* __restrict__ whw,     // (A)
    const float* __restrict__ whb,     // (1)
    const float* __restrict__ spatial, // (B, P, D)
    const float* __restrict__ s2,      // (B, D)
    const float* __restrict__ ht,      // (B, D)
    float* __restrict__ att_in)        // (B, D) = c_hat + ht
{
  const int b = blockIdx.x;
  const int lane = threadIdx.x & 31;
  const int wv = threadIdx.x >> 5;     // 8 waves
  __shared__ float zsh[PP + 1];
  __shared__ float alpha[PP + 1];
  const float* hb = hid + (long long)b * DD;

  for (int p = wv; p < PP + 1; p += 8) {
    const float* row = (p < PP) ? (vatt + ((long long)b * PP + p) * DD)
                                : (sen + (long long)b * DD);
    float part = 0.f;
    for (int d = lane; d < DD; d += 32)
      part += tanhf(row[d] + hb[d]) * whw[d];
    for (int off = 16; off > 0; off >>= 1)
      part += __shfl_down(part, off, 32);
    if (lane == 0) zsh[p] = part + whb[0];
  }
  __syncthreads();
  if (threadIdx.x == 0) {
    float mx = zsh[0];
    for (int p = 1; p < PP + 1; ++p) mx = fmaxf(mx, zsh[p]);
    float ssum = 0.f;
    for (int p = 0; p < PP + 1; ++p) { float e = expf(zsh[p] - mx); alpha[p] = e; ssum += e; }
    float inv = 1.f / ssum;
    for (int p = 0; p < PP + 1; ++p) alpha[p] *= inv;
  }
  __syncthreads();
  for (int d = threadIdx.x; d < DD; d += 256) {
    float acc = alpha[PP] * s2[(long long)b * DD + d];
    for (int p = 0; p < PP; ++p)
      acc += alpha[p] * spatial[((long long)b * PP + p) * DD + d];
    att_in[(long long)b * DD + d] = acc + ht[(long long)b * DD + d];
  }
}

// ---------------------------------------------------------------------------
// Build x_word = [emb[cap[b,t]] ; global_image[b]]  for t in 0..48
// ---------------------------------------------------------------------------
__global__ __launch_bounds__(256) void embed_kernel(
    const int* __restrict__ caps,      // (B, 50)
    const float* __restrict__ emb,     // (V, E)
    const float* __restrict__ gimg,    // (B, E)
    float* __restrict__ x_word)        // (B, 49, 1024)
{
  long long idx = (long long)blockIdx.x * 256 + threadIdx.x;
  if (idx >= (long long)BB * NS * 1024) return;
  int col = (int)(idx & 1023);
  long long bt = idx >> 10;
  int t = (int)(bt % NS);
  int b = (int)(bt / NS);
  float v;
  if (col < DD) v = emb[(long long)caps[b * TT + t] * DD + col];
  else          v = gimg[(long long)b * DD + (col - DD)];
  x_word[idx] = v;
}

__global__ __launch_bounds__(256) void zero_kernel(float* p, int n) {
  int i = blockIdx.x * 256 + threadIdx.x;
  if (i < n) p[i] = 0.f;
}

// ---------------------------------------------------------------------------
static inline void launch_gemm(hipStream_t s,
    const float* A0, int lda0, const float* W0, int ldw0, int K0,
    const float* A1, int lda1, const float* W1, int ldw1, int K1,
    const float* A2, int lda2, const float* W2, int ldw2, int K2,
    const float* b0, const float* b1,
    float* C, long long ldc, int M, int N, int act)
{
  dim3 grid((unsigned)((N + 255) / 256), (unsigned)(M / 32));
  wmma_gemm3<<<grid, dim3(256), 0, s>>>(A0, lda0, W0, ldw0, K0,
                                        A1, lda1, W1, ldw1, K1,
                                        A2, lda2, W2, ldw2, K2,
                                        b0, b1, C, ldc, N, act);
}

extern "C" void kernel_launch(void* const* d_in, const int* in_sizes, int n_in,
                              void* d_out, int out_size, void* d_ws, size_t ws_size,
                              hipStream_t stream)
{
  (void)in_sizes; (void)n_in; (void)out_size; (void)ws_size;
  const float* spatial = (const float*)d_in[0];
  const float* gimg    = (const float*)d_in[1];
  const int*   caps    = (const int*)d_in[2];
  // d_in[3] caption_lengths: identity no-op in reference
  const float* emb   = (const float*)d_in[4];
  const float* w_ih1 = (const float*)d_in[5];
  const float* w_hh1 = (const float*)d_in[6];
  const float* b_ih1 = (const float*)d_in[7];
  const float* b_hh1 = (const float*)d_in[8];
  const float* s_wx  = (const float*)d_in[9];
  const float* s_bx  = (const float*)d_in[10];
  const float* s_wh  = (const float*)d_in[11];
  const float* s_bh  = (const float*)d_in[12];
  const float* w_ih2 = (const float*)d_in[13];
  const float* w_hh2 = (const float*)d_in[14];
  const float* b_ih2 = (const float*)d_in[15];
  const float* b_hh2 = (const float*)d_in[16];
  const float* aff_s_w = (const float*)d_in[17];
  const float* aff_s_b = (const float*)d_in[18];
  const float* aff_h_w = (const float*)d_in[19];
  const float* aff_h_b = (const float*)d_in[20];
  const float* ws_w = (const float*)d_in[21];
  const float* ws_b = (const float*)d_in[22];
  const float* wg_w = (const float*)d_in[23];
  const float* wg_b = (const float*)d_in[24];
  const float* wv_w = (const float*)d_in[25];
  const float* wv_b = (const float*)d_in[26];
  const float* wh_w = (const float*)d_in[27];
  const float* wh_b = (const float*)d_in[28];
  const float* wp_w = (const float*)d_in[29];
  const float* wp_b = (const float*)d_in[30];
  const float* fc_w = (const float*)d_in[31];
  const float* fc_b = (const float*)d_in[32];
  float* out = (float*)d_out;

  // ---- workspace layout (floats) ----
  float* ws = (float*)d_ws;
  float* x_word = ws;  ws += (size_t)BB * NS * 1024;   // 6.42 M
  float* vatt   = ws;  ws += (size_t)BB * PP * DD;     // 3.21 M
  float* h1 = ws;      ws += BB * DD;
  float* m1 = ws;      ws += BB * DD;
  float* h2 = ws;      ws += BB * DD;
  float* m2 = ws;      ws += BB * DD;
  float* gates1 = ws;  ws += BB * 4 * DD;
  float* gates2 = ws;  ws += BB * 4 * DD;
  float* s_lin = ws;   ws += BB * DD;
  float* s_t   = ws;   ws += BB * DD;
  float* s2v   = ws;   ws += BB * DD;
  float* htv   = ws;   ws += BB * DD;
  float* hid   = ws;   ws += BB * DD;
  float* sen   = ws;   ws += BB * DD;
  float* att_in  = ws; ws += BB * DD;
  float* att_out = ws; ws += BB * DD;

  const int stateN = 4 * BB * DD;
  zero_kernel<<<(stateN + 255) / 256, 256, 0, stream>>>(h1, stateN); // h1,m1,h2,m2 contiguous

  {
    long long n = (long long)BB * NS * 1024;
    embed_kernel<<<(unsigned)((n + 255) / 256), 256, 0, stream>>>(caps, emb, gimg, x_word);
  }

  // visual_att = spatial @ wv_w^T + wv_b   (M = B*P = 6272)
  launch_gemm(stream, spatial, DD, wv_w, DD, DD,
              nullptr, 0, nullptr, 0, 0, nullptr, 0, nullptr, 0, 0,
              wv_b, nullptr, vatt, DD, BB * PP, DD, ACT_NONE);

  const long long ldx = (long long)NS * 1024;   // x_word row stride
  for (int t = 0; t < NS; ++t) {
    const float* x_t = x_word + (long long)t * 1024;

    // gates1 = b_ih1 + b_hh1 + [h2 | x_t] @ w_ih1^T + h1 @ w_hh1^T
    launch_gemm(stream,
                h2, DD,          w_ih1,        3 * DD, DD,        // cols 0..511 of w_ih1 rows
                x_t, (int)ldx,   w_ih1 + DD,   3 * DD, 2 * DD,    // cols 512..1535
                h1, DD,          w_hh1,        DD,     DD,
                b_ih1, b_hh1, gates1, 4 * DD, BB, 4 * DD, ACT_NONE);

    // s_lin = s_bx + s_bh + [h2 | x_t] @ s_wx^T + h1 @ s_wh^T
    launch_gemm(stream,
                h2, DD,          s_wx,         3 * DD, DD,
                x_t, (int)ldx,   s_wx + DD,    3 * DD, 2 * DD,
                h1, DD,          s_wh,         DD,     DD,
                s_bx, s_bh, s_lin, DD, BB, DD, ACT_NONE);

    // LSTM-1 activations + sentinel (writes new h1, m1, s_t)
    lstm_kernel<<<(BB * DD + 255) / 256, 256, 0, stream>>>(gates1, m1, h1, s_lin, s_t);

    // s2 = relu(s_t @ aff_s_w^T + aff_s_b);  ht = tanh(h1 @ aff_h_w^T + aff_h_b)
    launch_gemm(stream, s_t, DD, aff_s_w, DD, DD,
                nullptr, 0, nullptr, 0, 0, nullptr, 0, nullptr, 0, 0,
                aff_s_b, nullptr, s2v, DD, BB, DD, ACT_RELU);
    launch_gemm(stream, h1, DD, aff_h_w, DD, DD,
                nullptr, 0, nullptr, 0, 0, nullptr, 0, nullptr, 0, 0,
                aff_h_b, nullptr, htv, DD, BB, DD, ACT_TANH);

    // hid = ht @ wg_w^T + wg_b ; sen = s2 @ ws_w^T + ws_b
    launch_gemm(stream, htv, DD, wg_w, DD, DD,
                nullptr, 0, nullptr, 0, 0, nullptr, 0, nullptr, 0, 0,
                wg_b, nullptr, hid, DD, BB, DD, ACT_NONE);
    launch_gemm(stream, s2v, DD, ws_w, DD, DD,
                nullptr, 0, nullptr, 0, 0, nullptr, 0, nullptr, 0, 0,
                ws_b, nullptr, sen, DD, BB, DD, ACT_NONE);

    // attention: z, softmax, c_hat; att_in = c_hat + ht
    attn_kernel<<<BB, 256, 0, stream>>>(vatt, hid, sen, wh_w, wh_b,
                                        spatial, s2v, htv, att_in);

    // att_out = tanh(att_in @ wp_w^T + wp_b)
    launch_gemm(stream, att_in, DD, wp_w, DD, DD,
                nullptr, 0, nullptr, 0, 0, nullptr, 0, nullptr, 0, 0,
                wp_b, nullptr, att_out, DD, BB, DD, ACT_TANH);

    // gates2 = b_ih2 + b_hh2 + [att_out | h1] @ w_ih2^T + h2 @ w_hh2^T
    launch_gemm(stream,
                att_out, DD, w_ih2,      2 * DD, DD,
                h1, DD,      w_ih2 + DD, 2 * DD, DD,
                h2, DD,      w_hh2,      DD,     DD,
                b_ih2, b_hh2, gates2, 4 * DD, BB, 4 * DD, ACT_NONE);

    // LSTM-2 activations (writes new h2, m2)
    lstm_kernel<<<(BB * DD + 255) / 256, 256, 0, stream>>>(gates2, m2, h2, nullptr, nullptr);

    // preds[:, t, :] = h2 @ fc_w^T + fc_b  -> straight into d_out
    launch_gemm(stream, h2, DD, fc_w, DD, DD,
                nullptr, 0, nullptr, 0, 0, nullptr, 0, nullptr, 0, 0,
                fc_b, nullptr, out + (long long)t * VV, (long long)NS * VV,
                BB, VV, ACT_NONE);
  }
}